// LightGCN_28406913696113
// MI455X (gfx1250) — compile-verified
//
#include <hip/hip_runtime.h>
#include <hip/hip_bf16.h>

typedef __attribute__((ext_vector_type(2))) float v2f;
typedef __attribute__((ext_vector_type(8))) float v8f;

#define FD 64           // feature dim (reference uses 64)
#define DEG_EPS 1e-8f

// ---------------- elementwise helpers ----------------

__global__ void k_zero(float* __restrict__ p, long n) {
    long i = (long)blockIdx.x * blockDim.x + threadIdx.x;
    long stride = (long)gridDim.x * blockDim.x;
    for (; i < n; i += stride) p[i] = 0.0f;
}

__global__ void k_copy(float* __restrict__ dst, const float* __restrict__ src, long n) {
    long i = (long)blockIdx.x * blockDim.x + threadIdx.x;
    long stride = (long)gridDim.x * blockDim.x;
    for (; i < n; i += stride) dst[i] = src[i];
}

// deg[i] -> 1/(sqrt(deg[i]) + 1e-8), in place (applied to both deg_r and deg_c
// since they are contiguous in the workspace)
__global__ void k_invsqrt(float* __restrict__ d, long n) {
    long i = (long)blockIdx.x * blockDim.x + threadIdx.x;
    long stride = (long)gridDim.x * blockDim.x;
    for (; i < n; i += stride) d[i] = 1.0f / (sqrtf(d[i]) + DEG_EPS);
}

// ---------------- degrees ----------------

__global__ void k_degree(const int* __restrict__ row, const int* __restrict__ col,
                         const float* __restrict__ vals,
                         float* __restrict__ degr, float* __restrict__ degc, int E) {
    int i = blockIdx.x * blockDim.x + threadIdx.x;
    int stride = gridDim.x * blockDim.x;
    for (; i < E; i += stride) {
        float v = vals[i];
        atomicAdd(degr + row[i], v);
        atomicAdd(degc + col[i], v);
    }
}

// ---------------- SpMM: fout[row] += w * fin[col], one wave per edge ----------------
// All 32 lanes load the same row/col (cacheline broadcast); lane t owns the
// float2 at offset 2t of the 64-float feature row -> 256B coalesced read from
// L2-resident fin, 64 f32 atomic adds into L2-resident fout.

__global__ void k_spmm(const int* __restrict__ row, const int* __restrict__ col,
                       const float* __restrict__ vals,
                       const float* __restrict__ invr, const float* __restrict__ invc,
                       const float* __restrict__ fin, float* __restrict__ fout, int E) {
    int tid   = blockIdx.x * blockDim.x + threadIdx.x;
    int wave  = tid >> 5;
    int lane  = tid & 31;
    int nwaves = (gridDim.x * blockDim.x) >> 5;
    for (int e = wave; e < E; e += nwaves) {
        int r = row[e];
        int c = col[e];
        float w = vals[e] * invr[r] * invc[c];
        const float2* src = (const float2*)(fin + (long)c * FD);
        float2 f = src[lane];
        float* dst = fout + (long)r * FD + lane * 2;
        atomicAdd(dst,     w * f.x);
        atomicAdd(dst + 1, w * f.y);
    }
}

// ---------------- per-node L2 normalize + accumulate: acc += f/max(||f||,1e-12) ----------------
// One wave32 per node; lane t owns float2 at offset 2t. Wave reduction via shfl_xor.

__global__ void k_norm_acc(const float* __restrict__ f, float* __restrict__ acc, int N) {
    int tid  = blockIdx.x * blockDim.x + threadIdx.x;
    int wave = tid >> 5;
    int lane = tid & 31;
    if (wave >= N) return;
    const float2* fr = (const float2*)(f + (long)wave * FD);
    float2 v = fr[lane];
    float s = v.x * v.x + v.y * v.y;
    #pragma unroll
    for (int o = 16; o > 0; o >>= 1) s += __shfl_xor(s, o, 32);
    float inv = 1.0f / fmaxf(sqrtf(s), 1e-12f);
    float2* ar = (float2*)(acc + (long)wave * FD);
    float2 a = ar[lane];
    a.x += v.x * inv;
    a.y += v.y * inv;
    ar[lane] = a;
}

// ---------------- BPR loss with V_WMMA_F32_16X16X4_F32 ----------------
// One wave per tile of 16 batch elements. Computes the 16x16 products
// A_tile x Pos_tile^T and A_tile x Neg_tile^T accumulated over 16 chunks of
// K=4, then extracts the diagonal (the per-row dot products).
//
// ISA 16x4 f32 A layout: lanes 0-15 hold M=0..15 {K=0,K=1}; lanes 16-31 hold
// M=0..15 {K=2,K=3}. B (4x16): lanes 0-15 hold N=0..15 {K=0,K=1}; lanes 16-31
// N=0..15 {K=2,K=3}. So lane L loads row/col (L&15) at K-offset (L>>4)*2 for
// both operands. C/D f32 16x16: N=lane (lanes 0-15, M=c-index) and N=lane-16
// (lanes 16-31, M=c-index+8). Diagonal: i<8 -> lane i, c[i]; i>=8 -> lane i+16,
// c[i-8].
//
// rep here is the *unscaled* accumulator; node_rep = acc/4, so each dot gets a
// 1/16 factor folded in before the loss.

__global__ void k_bpr(const float* __restrict__ rep, const int* __restrict__ aid,
                      const int* __restrict__ pid, const int* __restrict__ nid,
                      float* __restrict__ out, int B, float invB) {
    int tid   = blockIdx.x * blockDim.x + threadIdx.x;
    int wave  = tid >> 5;
    int lane  = tid & 31;
    int tiles = (B + 15) >> 4;
    if (wave >= tiles) return;           // wave-uniform; surviving waves keep EXEC all-ones

    int r  = lane & 15;                  // batch row within tile
    int kh = (lane >> 4) * 2;            // K sub-offset: 0 or 2
    int b  = wave * 16 + r;
    int bc = (b < B) ? b : (B - 1);      // clamp (tail rows excluded from the sum below)
    long ar = (long)aid[bc] * FD;
    long pr = (long)pid[bc] * FD;
    long nr = (long)nid[bc] * FD;

    v8f cp = {0.f, 0.f, 0.f, 0.f, 0.f, 0.f, 0.f, 0.f};
    v8f cn = {0.f, 0.f, 0.f, 0.f, 0.f, 0.f, 0.f, 0.f};
    #pragma unroll
    for (int k0 = 0; k0 < FD; k0 += 4) {
        int k = k0 + kh;
        v2f A  = *(const v2f*)(rep + ar + k);
        v2f Bp = *(const v2f*)(rep + pr + k);
        v2f Bn = *(const v2f*)(rep + nr + k);
        cp = __builtin_amdgcn_wmma_f32_16x16x4_f32(false, A, false, Bp, (short)0, cp, false, false);
        cn = __builtin_amdgcn_wmma_f32_16x16x4_f32(false, A, false, Bn, (short)0, cn, false, false);
    }

    // diagonal extraction + loss
    int idx = (lane < 8) ? lane : ((lane >= 24) ? (lane - 24) : -1);
    int bi  = (lane < 8) ? (wave * 16 + lane)
                         : ((lane >= 24) ? (wave * 16 + lane - 16) : B);
    if (idx >= 0 && bi < B) {
        float dp = 0.f, dn = 0.f;
        #pragma unroll
        for (int j = 0; j < 8; ++j) {
            if (j == idx) { dp = cp[j]; dn = cn[j]; }
        }
        float x = (dp - dn) * 0.0625f;   // (acc_a . acc_p)/16 - (acc_a . acc_n)/16
        // -log_sigmoid(x) = softplus(-x), numerically stable:
        float l = (x > 0.f) ? log1pf(expf(-x)) : (-x + log1pf(expf(x)));
        atomicAdd(out, l * invB);
    }
}

// ---------------- launch ----------------

extern "C" void kernel_launch(void* const* d_in, const int* in_sizes, int n_in,
                              void* d_out, int out_size, void* d_ws, size_t ws_size,
                              hipStream_t stream) {
    const float* feat0 = (const float*)d_in[0];
    const int*   erow  = (const int*)d_in[1];
    const int*   ecol  = (const int*)d_in[2];
    const float* evals = (const float*)d_in[3];
    const int*   aid   = (const int*)d_in[4];
    const int*   pid   = (const int*)d_in[5];
    const int*   nid   = (const int*)d_in[6];

    const long N = (long)in_sizes[0] / FD;
    const int  E = in_sizes[1];
    const int  B = in_sizes[4];

    float* ws   = (float*)d_ws;
    float* degr = ws;                 // N
    float* degc = ws + N;             // N   (contiguous with degr)
    float* fA   = ws + 2 * N;         // N*FD
    float* fB   = fA + N * FD;        // N*FD
    float* acc  = fB + N * FD;        // N*FD
    float* out  = (float*)d_out;

    const int T = 256;
    const long ND = N * FD;

    // degrees -> inverse sqrt factors
    k_zero<<<(int)((2 * N + T - 1) / T), T, 0, stream>>>(degr, 2 * N);
    k_zero<<<1, 32, 0, stream>>>(out, 1);
    k_degree<<<2048, T, 0, stream>>>(erow, ecol, evals, degr, degc, E);
    k_invsqrt<<<(int)((2 * N + T - 1) / T), T, 0, stream>>>(degr, 2 * N);

    // acc starts as the layer-0 (unnormalized) term
    k_copy<<<(int)((ND + T - 1) / T), T, 0, stream>>>(acc, feat0, ND);

    // 3 propagation layers (ping-pong fA/fB; features stay unnormalized)
    const float* fin = feat0;
    float* fout = fA;
    for (int layer = 0; layer < 3; ++layer) {
        k_zero<<<(int)((ND + T - 1) / T), T, 0, stream>>>(fout, ND);
        k_spmm<<<2048, T, 0, stream>>>(erow, ecol, evals, degr, degc, fin, fout, E);
        int nblk = (int)((N * 32 + T - 1) / T);
        k_norm_acc<<<nblk, T, 0, stream>>>(fout, acc, (int)N);
        fin  = fout;
        fout = (fout == fA) ? fB : fA;
    }

    // BPR loss via WMMA tiles
    int tiles = (B + 15) / 16;
    int bprBlocks = (tiles * 32 + T - 1) / T;
    k_bpr<<<bprBlocks, T, 0, stream>>>(acc, aid, pid, nid, out, B, 1.0f / (float)B);
}